// FSAM_70789650972916
// MI455X (gfx1250) — compile-verified
//
#include <hip/hip_runtime.h>
#include <math.h>

#define MTOT 65536           // 16*64*64
#define NMF_ITERS 50
#define EPSF 1.1920929e-07f

typedef __attribute__((ext_vector_type(2))) float v2f;
typedef __attribute__((ext_vector_type(8))) float v8f;
typedef unsigned int u32x4 __attribute__((ext_vector_type(4)));
typedef int i32x4 __attribute__((ext_vector_type(4)));
typedef int i32x8 __attribute__((ext_vector_type(8)));

static __device__ __forceinline__ v8f wmma4(v2f a, v2f b, v8f c) {
    // D(16x16,f32) = A(16x4,f32) * B(4x16,f32) + C
    return __builtin_amdgcn_wmma_f32_16x16x4_f32(
        false, a, false, b, (short)0, c, false, false);
}

static __device__ __forceinline__ v8f vzero8() {
    v8f z = {0.f,0.f,0.f,0.f,0.f,0.f,0.f,0.f};
    return z;
}

static __device__ __forceinline__ float u01(unsigned x) {
    x ^= x >> 17; x *= 0xed5ad4bbu; x ^= x >> 11; x *= 0xac4c1b51u;
    x ^= x >> 15; x *= 0x31848babu; x ^= x >> 14;
    return (float)(x >> 8) * (1.0f / 16777216.0f);
}

// ---------------------------------------------------------------------------
// Tensor Data Mover: DMA a rows x cols fp32 tile (row-major, stride == cols)
// from global memory into LDS. D# built per CDNA5 ISA ch.8 (2-D tensor,
// groups 2/3 zero). Issue from ONE wave, then s_wait_tensorcnt + barrier.
// This toolchain uses the 6-arg builtin (extra zero-filled i32x8 group).
// ---------------------------------------------------------------------------
static __device__ __forceinline__ void tdm_load_2d_f32(
    const float* gsrc, void* lds_dst, unsigned rows, unsigned cols)
{
    unsigned long long ga = (unsigned long long)(uintptr_t)gsrc;
    unsigned lds_off = (unsigned)(uintptr_t)lds_dst;   // addr[31:0] = LDS offset

    u32x4 g0;
    g0[0] = 1u;                                     // count=1 (valid descriptor)
    g0[1] = lds_off;                                // lds_addr (bytes)
    g0[2] = (unsigned)(ga & 0xFFFFFFFFu);           // global_addr[31:0]
    g0[3] = (unsigned)((ga >> 32) & 0x01FFFFFFu)    // global_addr[56:32]
          | (2u << 30);                             // type = 2 ("image")

    i32x8 g1;
    g1[0] = (int)(2u << 16);                        // data_size = 2 (4 bytes)
    g1[1] = (int)((cols & 0xFFFFu) << 16);          // tensor_dim0[15:0]
    g1[2] = (int)((cols >> 16) & 0xFFFFu)           // tensor_dim0[31:16]
          | (int)((rows & 0xFFFFu) << 16);          // tensor_dim1[15:0]
    g1[3] = (int)((rows >> 16) & 0xFFFFu)           // tensor_dim1[31:16]
          | (int)((cols & 0xFFFFu) << 16);          // tile_dim0 = cols
    g1[4] = (int)(rows & 0xFFFFu);                  // tile_dim1 = rows, tile_dim2=0
    g1[5] = (int)cols;                              // tensor_dim0_stride[31:0]
    g1[6] = 0;                                      // stride hi, dim1_stride lo
    g1[7] = 0;

    i32x4 z4 = {0, 0, 0, 0};                        // groups 2/3: 2-D mode
    i32x8 z8 = {0, 0, 0, 0, 0, 0, 0, 0};            // trailing group (unused)
    __builtin_amdgcn_tensor_load_to_lds(g0, g1, z4, z4, z8, 0);
}

// ---------------------------------------------------------------------------
// Vst = relu(w_pre * eps + b_pre), plus sum(Vst) for NMF init scale.
// grid: 1024 blocks x 128 thr (4 waves); wave -> one 16-wide m strip (64x16).
// ---------------------------------------------------------------------------
__global__ void __launch_bounds__(128) k_preconv(
    const float* __restrict__ eps, const float* __restrict__ w_pre,
    const float* __restrict__ b_pre, float* __restrict__ Vst,
    float* __restrict__ sumV)
{
    __shared__ float lw[64 * 64];
    __shared__ float red[128];
    const int tid = threadIdx.x;

    if ((tid >> 5) == 0) {                          // wave 0 drives the TDM
        tdm_load_2d_f32(w_pre, lw, 64, 64);
        __builtin_amdgcn_s_wait_tensorcnt(0);
    }
    __syncthreads();

    const int lane = tid & 31;
    const int l16  = lane & 15;
    const int kh   = (lane >> 4) << 1;   // K offset: 0 or 2
    const int h8   = (lane >> 4) << 3;   // row offset: 0 or 8
    const int m0   = (blockIdx.x * 4 + (tid >> 5)) * 16;

    v8f acc[4];
    for (int ct = 0; ct < 4; ++ct) {
        acc[ct] = vzero8();
        for (int i = 0; i < 8; ++i) acc[ct][i] = b_pre[ct * 16 + i + h8];
    }

    for (int k0 = 0; k0 < 64; k0 += 4) {
        v2f b;
        b.x = eps[(size_t)(k0 + kh)     * MTOT + m0 + l16];
        b.y = eps[(size_t)(k0 + kh + 1) * MTOT + m0 + l16];
        for (int ct = 0; ct < 4; ++ct) {
            v2f a;
            a.x = lw[(ct * 16 + l16) * 64 + k0 + kh];
            a.y = lw[(ct * 16 + l16) * 64 + k0 + kh + 1];
            acc[ct] = wmma4(a, b, acc[ct]);
        }
    }

    float s = 0.f;
    for (int ct = 0; ct < 4; ++ct)
        for (int i = 0; i < 8; ++i) {
            float v = acc[ct][i];
            v = v > 0.f ? v : 0.f;
            Vst[(size_t)(ct * 16 + i + h8) * MTOT + m0 + l16] = v;
            s += v;
        }

    red[tid] = s;
    __syncthreads();
    for (int off = 64; off; off >>= 1) {
        if (tid < off) red[tid] += red[tid + off];
        __syncthreads();
    }
    if (tid == 0) unsafeAtomicAdd(sumV, red[0]);
}

__global__ void k_zero2(float* a, float* b) {
    if (threadIdx.x == 0) { *a = 0.f; *b = 0.f; }
}

// Init Wp[64x16] (cols 0..7 uniform*avg, 8..15 zero) and Hp[16x65536] likewise.
__global__ void __launch_bounds__(256) k_init(
    float* __restrict__ Hp, float* __restrict__ Wp, const float* __restrict__ sumV)
{
    const float avg = sqrtf(sumV[0] * (1.0f / (64.0f * 65536.0f)) * 0.125f);
    const unsigned idx = blockIdx.x * 256u + threadIdx.x;
    if (idx < 16u * 65536u) {
        unsigned r = idx >> 16;
        Hp[idx] = (r < 8u) ? avg * u01(idx * 2654435761u + 12345u) : 0.f;
    }
    if (idx < 1024u) {
        unsigned rr = idx & 15u;
        Wp[idx] = (rr < 8u) ? avg * u01(idx * 2246822519u + 999u) : 0.f;
    }
}

// S = Wp^T * Wp  (16x16). One wave; A and B operands are identical lane data.
__global__ void k_wtw(const float* __restrict__ Wp, float* __restrict__ S)
{
    const int lane = threadIdx.x & 31;
    const int l16 = lane & 15, kh = (lane >> 4) << 1, h8 = (lane >> 4) << 3;
    v8f acc = vzero8();
    for (int k0 = 0; k0 < 64; k0 += 4) {
        v2f ab;
        ab.x = Wp[(k0 + kh)     * 16 + l16];
        ab.y = Wp[(k0 + kh + 1) * 16 + l16];
        acc = wmma4(ab, ab, acc);
    }
    for (int i = 0; i < 8; ++i) S[(i + h8) * 16 + l16] = acc[i];
}

// H *= (Wp^T V) / (S*H + eps), per 16-wide m strip (rows 8..15 stay zero).
__global__ void __launch_bounds__(128) k_h_update(
    const float* __restrict__ V, const float* __restrict__ Wp,
    const float* __restrict__ S, float* __restrict__ Hp)
{
    __shared__ float lwp[64 * 16];
    __shared__ float ls[256];
    const int tid = threadIdx.x;

    if ((tid >> 5) == 0) {                          // TDM-stage Wp and S
        tdm_load_2d_f32(Wp, lwp, 64, 16);
        tdm_load_2d_f32(S, ls, 16, 16);
        __builtin_amdgcn_s_wait_tensorcnt(0);
    }
    __syncthreads();

    const int lane = tid & 31;
    const int l16 = lane & 15, kh = (lane >> 4) << 1, h8 = (lane >> 4) << 3;
    const int m0 = (blockIdx.x * 4 + (tid >> 5)) * 16;

    v8f accW = vzero8();                       // (Wp^T V) tile
    for (int k0 = 0; k0 < 64; k0 += 4) {
        v2f a, b;
        a.x = lwp[(k0 + kh)     * 16 + l16];   // A[r][t] = Wp[t][r]
        a.y = lwp[(k0 + kh + 1) * 16 + l16];
        b.x = V[(size_t)(k0 + kh)     * MTOT + m0 + l16];
        b.y = V[(size_t)(k0 + kh + 1) * MTOT + m0 + l16];
        accW = wmma4(a, b, accW);
    }
    v8f accS = vzero8();                       // (S*H) tile
    for (int k0 = 0; k0 < 16; k0 += 4) {
        v2f a, b;
        a.x = ls[l16 * 16 + k0 + kh];
        a.y = ls[l16 * 16 + k0 + kh + 1];
        b.x = Hp[(size_t)(k0 + kh)     * MTOT + m0 + l16];
        b.y = Hp[(size_t)(k0 + kh + 1) * MTOT + m0 + l16];
        accS = wmma4(a, b, accS);
    }
    for (int i = 0; i < 8; ++i) {
        size_t idx = (size_t)(i + h8) * MTOT + m0 + l16;
        float h = Hp[idx];
        Hp[idx] = h * accW[i] / (accS[i] + EPSF);
    }
}

__global__ void k_zero_nd(float* VHt, float* G) {
    int i = threadIdx.x;
    for (int j = i; j < 1024; j += 256) VHt[j] = 0.f;
    if (i < 256) G[i] = 0.f;
}

// VHt = V*Hp^T (64x16) and G = Hp*Hp^T (16x16), reduced over m with atomics.
// grid: 16 blocks x 128 thr = 64 waves; each wave handles 64 m-tiles.
__global__ void __launch_bounds__(128) k_w_numden(
    const float* __restrict__ V, const float* __restrict__ Hp,
    float* __restrict__ VHt, float* __restrict__ G)
{
    const int tid = threadIdx.x;
    const int lane = tid & 31;
    const int l16 = lane & 15, kh = (lane >> 4) << 1, h8 = (lane >> 4) << 3;
    const int wv = blockIdx.x * 4 + (tid >> 5);       // 0..63

    v8f acc[4];
    for (int ct = 0; ct < 4; ++ct) acc[ct] = vzero8();
    v8f accG = vzero8();

    for (int t = 0; t < 64; ++t) {
        const int m0 = (wv * 64 + t) * 16;
        {   // pull the next m-tile toward the WGP (global_prefetch_b8)
            const int mn = (wv * 64 + ((t + 1) & 63)) * 16;
            __builtin_prefetch(&Hp[(size_t)l16 * MTOT + mn], 0, 1);
            __builtin_prefetch(&V[(size_t)l16 * MTOT + mn], 0, 1);
        }
        for (int ks = 0; ks < 16; ks += 4) {
            const int km = m0 + ks;                   // K runs along m
            v2f b;                                    // B[k][r] = Hp[r][km+k]
            b.x = Hp[(size_t)l16 * MTOT + km + kh];
            b.y = Hp[(size_t)l16 * MTOT + km + kh + 1];
            accG = wmma4(b, b, accG);                 // A for G is same lane data
            for (int ct = 0; ct < 4; ++ct) {
                v2f a;                                // A[c][k] = V[c][km+k]
                a.x = V[(size_t)(ct * 16 + l16) * MTOT + km + kh];
                a.y = V[(size_t)(ct * 16 + l16) * MTOT + km + kh + 1];
                acc[ct] = wmma4(a, b, acc[ct]);
            }
        }
    }
    for (int ct = 0; ct < 4; ++ct)
        for (int i = 0; i < 8; ++i)
            unsafeAtomicAdd(&VHt[(ct * 16 + i + h8) * 16 + l16], acc[ct][i]);
    for (int i = 0; i < 8; ++i)
        unsafeAtomicAdd(&G[(i + h8) * 16 + l16], accG[i]);
}

// W *= VHt / (Wp*G + eps). One block, 4 waves (one c-tile each).
__global__ void __launch_bounds__(128) k_w_update(
    float* __restrict__ Wp, const float* __restrict__ VHt,
    const float* __restrict__ G)
{
    __shared__ float lg[256];
    const int tid = threadIdx.x;
    for (int i = tid; i < 256; i += 128) lg[i] = G[i];
    __syncthreads();
    const int lane = tid & 31;
    const int l16 = lane & 15, kh = (lane >> 4) << 1, h8 = (lane >> 4) << 3;
    const int row0 = (tid >> 5) * 16;

    v8f acc = vzero8();
    for (int k0 = 0; k0 < 16; k0 += 4) {
        v2f a, b;
        a.x = Wp[(row0 + l16) * 16 + k0 + kh];
        a.y = Wp[(row0 + l16) * 16 + k0 + kh + 1];
        b.x = lg[(k0 + kh)     * 16 + l16];
        b.y = lg[(k0 + kh + 1) * 16 + l16];
        acc = wmma4(a, b, acc);
    }
    for (int i = 0; i < 8; ++i) {
        int t = row0 + i + h8;
        float w = Wp[t * 16 + l16];
        Wp[t * 16 + l16] = w * VHt[t * 16 + l16] / (acc[i] + EPSF);
    }
}

// Fused epilogue per 16-wide m strip:
//   Vhat = Wp*Hp ; err += |Vst-Vhat|^2 ; y = relu(w1*Vhat) ; out = w2*y
// Vhat and y staged through per-wave LDS to convert D-layout -> B-layout.
__global__ void __launch_bounds__(128) k_fused_post(
    const float* __restrict__ Vst, const float* __restrict__ Hp,
    const float* __restrict__ Wp, const float* __restrict__ w1,
    const float* __restrict__ w2, float* __restrict__ out,
    float* __restrict__ err2)
{
    __shared__ float lw1[4096];
    __shared__ float lw2[4096];
    __shared__ float stage[4][1024];
    __shared__ float red[128];
    const int tid = threadIdx.x;

    if ((tid >> 5) == 0) {                          // TDM-stage both weights
        tdm_load_2d_f32(w1, lw1, 64, 64);
        tdm_load_2d_f32(w2, lw2, 64, 64);
        __builtin_amdgcn_s_wait_tensorcnt(0);
    }
    __syncthreads();

    const int lane = tid & 31;
    const int l16 = lane & 15, kh = (lane >> 4) << 1, h8 = (lane >> 4) << 3;
    const int wave = tid >> 5;
    const int m0 = (blockIdx.x * 4 + wave) * 16;
    float* st = stage[wave];

    // Vhat = Wp(64x16) * Hp_tile(16x16)
    v8f acc[4];
    for (int ct = 0; ct < 4; ++ct) acc[ct] = vzero8();
    for (int k0 = 0; k0 < 16; k0 += 4) {
        v2f b;
        b.x = Hp[(size_t)(k0 + kh)     * MTOT + m0 + l16];
        b.y = Hp[(size_t)(k0 + kh + 1) * MTOT + m0 + l16];
        for (int ct = 0; ct < 4; ++ct) {
            v2f a;
            a.x = Wp[(ct * 16 + l16) * 16 + k0 + kh];
            a.y = Wp[(ct * 16 + l16) * 16 + k0 + kh + 1];
            acc[ct] = wmma4(a, b, acc[ct]);
        }
    }
    float s = 0.f;
    for (int ct = 0; ct < 4; ++ct)
        for (int i = 0; i < 8; ++i) {
            int r = ct * 16 + i + h8;
            float vh = acc[ct][i];
            float d = Vst[(size_t)r * MTOT + m0 + l16] - vh;
            s += d * d;
            st[r * 16 + l16] = vh;
        }

    // y = relu(w1 * Vhat)
    v8f acc2[4];
    for (int ct = 0; ct < 4; ++ct) acc2[ct] = vzero8();
    for (int k0 = 0; k0 < 64; k0 += 4) {
        v2f b;
        b.x = st[(k0 + kh)     * 16 + l16];
        b.y = st[(k0 + kh + 1) * 16 + l16];
        for (int ct = 0; ct < 4; ++ct) {
            v2f a;
            a.x = lw1[(ct * 16 + l16) * 64 + k0 + kh];
            a.y = lw1[(ct * 16 + l16) * 64 + k0 + kh + 1];
            acc2[ct] = wmma4(a, b, acc2[ct]);
        }
    }
    for (int ct = 0; ct < 4; ++ct)
        for (int i = 0; i < 8; ++i) {
            float v = acc2[ct][i];
            st[(ct * 16 + i + h8) * 16 + l16] = v > 0.f ? v : 0.f;
        }

    // eps_hat = w2 * y
    v8f acc3[4];
    for (int ct = 0; ct < 4; ++ct) acc3[ct] = vzero8();
    for (int k0 = 0; k0 < 64; k0 += 4) {
        v2f b;
        b.x = st[(k0 + kh)     * 16 + l16];
        b.y = st[(k0 + kh + 1) * 16 + l16];
        for (int ct = 0; ct < 4; ++ct) {
            v2f a;
            a.x = lw2[(ct * 16 + l16) * 64 + k0 + kh];
            a.y = lw2[(ct * 16 + l16) * 64 + k0 + kh + 1];
            acc3[ct] = wmma4(a, b, acc3[ct]);
        }
    }
    for (int ct = 0; ct < 4; ++ct)
        for (int i = 0; i < 8; ++i)
            out[(size_t)(ct * 16 + i + h8) * MTOT + m0 + l16] = acc3[ct][i];

    red[tid] = s;
    __syncthreads();
    for (int off = 64; off; off >>= 1) {
        if (tid < off) red[tid] += red[tid + off];
        __syncthreads();
    }
    if (tid == 0) unsafeAtomicAdd(err2, red[0]);
}

__global__ void k_finalize(const float* __restrict__ err2, float* __restrict__ out) {
    if (threadIdx.x == 0) out[4194304] = sqrtf(err2[0]);
}

// ---------------------------------------------------------------------------
extern "C" void kernel_launch(void* const* d_in, const int* in_sizes, int n_in,
                              void* d_out, int out_size, void* d_ws, size_t ws_size,
                              hipStream_t stream)
{
    const float* eps     = (const float*)d_in[0];
    const float* w_pre   = (const float*)d_in[1];
    const float* b_pre   = (const float*)d_in[2];
    const float* w_post1 = (const float*)d_in[3];
    const float* w_post2 = (const float*)d_in[4];
    float* out = (float*)d_out;

    float* Vst  = (float*)d_ws;                 // 64 x 65536
    float* Hp   = Vst + (size_t)64 * MTOT;      // 16 x 65536 (rows 8..15 zero)
    float* Wp   = Hp + (size_t)16 * MTOT;       // 64 x 16    (cols 8..15 zero)
    float* S    = Wp + 1024;                    // 16 x 16
    float* VHt  = S + 256;                      // 64 x 16
    float* G    = VHt + 1024;                   // 16 x 16
    float* sumV = G + 256;
    float* err2 = sumV + 1;

    k_zero2<<<1, 32, 0, stream>>>(sumV, err2);
    k_preconv<<<1024, 128, 0, stream>>>(eps, w_pre, b_pre, Vst, sumV);
    k_init<<<4096, 256, 0, stream>>>(Hp, Wp, sumV);

    for (int it = 0; it < NMF_ITERS; ++it) {
        k_wtw<<<1, 32, 0, stream>>>(Wp, S);
        k_h_update<<<1024, 128, 0, stream>>>(Vst, Wp, S, Hp);
        k_zero_nd<<<1, 256, 0, stream>>>(VHt, G);
        k_w_numden<<<16, 128, 0, stream>>>(Vst, Hp, VHt, G);
        k_w_update<<<1, 128, 0, stream>>>(Wp, VHt, G);
    }

    k_fused_post<<<1024, 128, 0, stream>>>(Vst, Hp, Wp, w_post1, w_post2, out, err2);
    k_finalize<<<1, 32, 0, stream>>>(err2, out);
}